// rsam_22608707846224
// MI455X (gfx1250) — compile-verified
//
#include <hip/hip_runtime.h>

// RSAM-GCN inference for MI455X (gfx1250, wave32).
// fp32 WMMA (V_WMMA_F32_16X16X4_F32) for all dense ops; paired-K LDS layouts
// (fragment = one ds_load_b64, pairs fuse to ds_load_2addr_b64); 2-deep
// software-pipelined K loops (two fragment register sets, loads a full WMMA
// group ahead of their consumers); 4-wide / 2x2 register tiling so one
// fragment feeds 4 WMMAs; f32 global atomics + global_prefetch for the graph
// scatter; fused epilogue (softmax-mix combine + self-loop + bias + BN + ReLU).

#define N_NODES 50000
#define N_EDGES 800000
#define D_IN    128
#define D_H     128
#define D_OUTF  64
#define BN_EPS  1e-5f
#define GRAM_CHUNK 128   // 128 nodes * 128 feats * 4B = 64KB LDS

typedef float v2f __attribute__((ext_vector_type(2)));
typedef float v8f __attribute__((ext_vector_type(8)));

static __device__ __forceinline__ v8f wmma_f32(v2f a, v2f b, v8f c) {
  // D = A(16x4,f32) * B(4x16,f32) + C(16x16,f32)
  return __builtin_amdgcn_wmma_f32_16x16x4_f32(
      /*neg_a=*/false, a, /*neg_b=*/false, b,
      /*c_mod=*/(short)0, c, /*reuse_a=*/false, /*reuse_b=*/false);
}

// ---------------------------------------------------------------- utilities
__global__ void fill_kernel(float* __restrict__ p, float val, long n) {
  long i = (long)blockIdx.x * blockDim.x + threadIdx.x;
  long stride = (long)gridDim.x * blockDim.x;
  for (; i < n; i += stride) p[i] = val;
}

__global__ void deg_kernel(const int* __restrict__ col, float* __restrict__ deg, int E) {
  int e = blockIdx.x * blockDim.x + threadIdx.x;
  if (e < E) atomicAdd(&deg[col[e]], 1.0f);
}

__global__ void norm_kernel(float* __restrict__ dis, float* __restrict__ dis2, int n) {
  int i = blockIdx.x * blockDim.x + threadIdx.x;
  if (i < n) {
    float d = dis[i];                       // holds degree (incl. self loop)
    float r = d > 0.0f ? rsqrtf(d) : 0.0f;
    dis[i]  = r;                            // deg^{-1/2}
    dis2[i] = r * r;                        // 1/deg  (self-loop weight)
  }
}

__global__ void ew_kernel(const int* __restrict__ row, const int* __restrict__ col,
                          const float* __restrict__ dis, float* __restrict__ ew, int E) {
  int e = blockIdx.x * blockDim.x + threadIdx.x;
  if (e < E) ew[e] = dis[row[e]] * dis[col[e]];
}

// ------------------------------------------------------------ graph scatter
// One wave per edge: agg[col] += ew * xp[row].  One vector load per lane
// (float4 for D=128, float2 for D=64) + D/32 f32 atomics.  Prefetch the
// source row of the edge 32 ahead (lanes spread to cover the whole row).
__global__ __launch_bounds__(256) void scatter_kernel(
    const int* __restrict__ rowi, const int* __restrict__ coli,
    const float* __restrict__ ew, const float* __restrict__ X,
    float* __restrict__ AGG, int E, int D)
{
  int gwid = (blockIdx.x * 256 + threadIdx.x) >> 5;   // global wave id (wave32)
  int lane = threadIdx.x & 31;
  if (gwid >= E) return;
  if (gwid + 32 < E) {
    int ps = rowi[gwid + 32];
    __builtin_prefetch(X + (long)ps * D + lane * (D >> 5), 0, 0);
  }
  int s = rowi[gwid], d = coli[gwid];
  float w = ew[gwid];
  const float* xs = X + (long)s * D;
  float* ad = AGG + (long)d * D;
  if (D == 128) {
    const float4 xv = *(const float4*)(xs + lane * 4);
    atomicAdd(&ad[lane * 4 + 0], w * xv.x);
    atomicAdd(&ad[lane * 4 + 1], w * xv.y);
    atomicAdd(&ad[lane * 4 + 2], w * xv.z);
    atomicAdd(&ad[lane * 4 + 3], w * xv.w);
  } else {
    const float2 xv = *(const float2*)(xs + lane * 2);
    atomicAdd(&ad[lane * 2 + 0], w * xv.x);
    atomicAdd(&ad[lane * 2 + 1], w * xv.y);
  }
}

// ------------------------------------------------------------- GEMM: h @ W
// 256 threads = 8 waves; each block covers 128 rows (8 M-tiles of 16).
// B staged in LDS with paired-K layout [(K/2)][NOUT][2] (fragment = one
// ds_load_b64).  4-wide N-tiling: one A fragment feeds 4 WMMAs.  2-deep
// pipelined K loop: fragment loads run a full WMMA group ahead.
__global__ __launch_bounds__(256) void gemm_wmma(
    const float* __restrict__ A, const float* __restrict__ B,
    float* __restrict__ C, int M, int K, int NOUT)
{
  extern __shared__ float sB[];              // K * NOUT floats, paired-K
  for (int i = threadIdx.x; i < K * NOUT; i += 256) {
    int k = i / NOUT, c = i - k * NOUT;
    sB[(k >> 1) * (2 * NOUT) + c * 2 + (k & 1)] = B[i];
  }
  __syncthreads();
  const int wave = threadIdx.x >> 5;
  const int lane = threadIdx.x & 31;
  const int r  = lane & 15;
  const int hi = lane >> 4;
  const int m0 = blockIdx.x * 128 + wave * 16;
  if (m0 >= M) return;                       // wave-uniform: EXEC stays full
  int ar = m0 + r; if (ar >= M) ar = M - 1;  // clamp tail loads (stores masked)
  const float* Arow = A + (long)ar * K + 2 * hi;
  for (int n0 = 0; n0 < NOUT; n0 += 64) {
    v8f acc0 = {}, acc1 = {}, acc2 = {}, acc3 = {};
    const float* pb = sB + hi * (2 * NOUT) + (n0 + r) * 2;
    // stage A (k=0)
    v2f aA  = *(const v2f*)(Arow);
    v2f b0A = *(const v2f*)(pb);
    v2f b1A = *(const v2f*)(pb + 32);
    v2f b2A = *(const v2f*)(pb + 64);
    v2f b3A = *(const v2f*)(pb + 96);
    // stage B (k=4)
    const float* pbl = pb + 4 * NOUT;
    v2f aB  = *(const v2f*)(Arow + 4);
    v2f b0B = *(const v2f*)(pbl);
    v2f b1B = *(const v2f*)(pbl + 32);
    v2f b2B = *(const v2f*)(pbl + 64);
    v2f b3B = *(const v2f*)(pbl + 96);
    pbl += 4 * NOUT;
    for (int k0 = 8; k0 < K; k0 += 8) {
      acc0 = wmma_f32(aA, b0A, acc0);
      acc1 = wmma_f32(aA, b1A, acc1);
      acc2 = wmma_f32(aA, b2A, acc2);
      acc3 = wmma_f32(aA, b3A, acc3);
      aA  = *(const v2f*)(Arow + k0);
      b0A = *(const v2f*)(pbl);
      b1A = *(const v2f*)(pbl + 32);
      b2A = *(const v2f*)(pbl + 64);
      b3A = *(const v2f*)(pbl + 96);
      pbl += 4 * NOUT;
      acc0 = wmma_f32(aB, b0B, acc0);
      acc1 = wmma_f32(aB, b1B, acc1);
      acc2 = wmma_f32(aB, b2B, acc2);
      acc3 = wmma_f32(aB, b3B, acc3);
      aB  = *(const v2f*)(Arow + k0 + 4);
      b0B = *(const v2f*)(pbl);
      b1B = *(const v2f*)(pbl + 32);
      b2B = *(const v2f*)(pbl + 64);
      b3B = *(const v2f*)(pbl + 96);
      pbl += 4 * NOUT;
    }
    acc0 = wmma_f32(aA, b0A, acc0);
    acc1 = wmma_f32(aA, b1A, acc1);
    acc2 = wmma_f32(aA, b2A, acc2);
    acc3 = wmma_f32(aA, b3A, acc3);
    acc0 = wmma_f32(aB, b0B, acc0);
    acc1 = wmma_f32(aB, b1B, acc1);
    acc2 = wmma_f32(aB, b2B, acc2);
    acc3 = wmma_f32(aB, b3B, acc3);
#pragma unroll
    for (int v = 0; v < 8; ++v) {
      int rr = m0 + v + hi * 8;
      if (rr < M) {
        float* cr = C + (long)rr * NOUT + n0 + r;
        cr[0]  = acc0[v];
        cr[16] = acc1[v];
        cr[32] = acc2[v];
        cr[48] = acc3[v];
      }
    }
  }
}

// ---------------------------------------------------------- Gram: xp^T @ xp
// Block stages a GRAM_CHUNK x D node slab in LDS (paired-K, zero-padded tail
// keeps EXEC all-ones).  2x2 block tiling (32x32 output block), 2-deep
// pipelined K loop; atomics into G.
__global__ __launch_bounds__(256) void gram_wmma(
    const float* __restrict__ X, float* __restrict__ G, int M, int D)
{
  extern __shared__ float sX[];              // GRAM_CHUNK * D floats, paired-K
  const int k_begin = blockIdx.x * GRAM_CHUNK;
  const int cnt = min(GRAM_CHUNK, M - k_begin);
  for (int i = threadIdx.x; i < GRAM_CHUNK * D; i += 256) {
    int k = i / D, f = i - k * D;
    float val = (k < cnt) ? X[(long)(k_begin + k) * D + f] : 0.0f;
    sX[(k >> 1) * (2 * D) + f * 2 + (k & 1)] = val;
  }
  __syncthreads();
  const int wave = threadIdx.x >> 5;
  const int lane = threadIdx.x & 31;
  const int r  = lane & 15;
  const int hi = lane >> 4;
  const int bdim = D >> 5;                   // 32x32 blocks
  for (int t = wave; t < bdim * bdim; t += 8) {
    const int f0 = (t / bdim) * 32;          // A = xp^T : A[f][k] = xp[k][f]
    const int g0 = (t % bdim) * 32;
    const float* pa = sX + hi * (2 * D) + (f0 + r) * 2;
    const float* pb = sX + hi * (2 * D) + (g0 + r) * 2;
    v8f acc00 = {}, acc01 = {}, acc10 = {}, acc11 = {};
    // stage A (k=0)
    v2f a0A = *(const v2f*)(pa);
    v2f a1A = *(const v2f*)(pa + 32);
    v2f b0A = *(const v2f*)(pb);
    v2f b1A = *(const v2f*)(pb + 32);
    pa += 4 * D; pb += 4 * D;
    // stage B (k=4)
    v2f a0B = *(const v2f*)(pa);
    v2f a1B = *(const v2f*)(pa + 32);
    v2f b0B = *(const v2f*)(pb);
    v2f b1B = *(const v2f*)(pb + 32);
    pa += 4 * D; pb += 4 * D;
    for (int k0 = 8; k0 < GRAM_CHUNK; k0 += 8) {
      acc00 = wmma_f32(a0A, b0A, acc00);
      acc01 = wmma_f32(a0A, b1A, acc01);
      acc10 = wmma_f32(a1A, b0A, acc10);
      acc11 = wmma_f32(a1A, b1A, acc11);
      a0A = *(const v2f*)(pa);
      a1A = *(const v2f*)(pa + 32);
      b0A = *(const v2f*)(pb);
      b1A = *(const v2f*)(pb + 32);
      pa += 4 * D; pb += 4 * D;
      acc00 = wmma_f32(a0B, b0B, acc00);
      acc01 = wmma_f32(a0B, b1B, acc01);
      acc10 = wmma_f32(a1B, b0B, acc10);
      acc11 = wmma_f32(a1B, b1B, acc11);
      a0B = *(const v2f*)(pa);
      a1B = *(const v2f*)(pa + 32);
      b0B = *(const v2f*)(pb);
      b1B = *(const v2f*)(pb + 32);
      pa += 4 * D; pb += 4 * D;
    }
    acc00 = wmma_f32(a0A, b0A, acc00);
    acc01 = wmma_f32(a0A, b1A, acc01);
    acc10 = wmma_f32(a1A, b0A, acc10);
    acc11 = wmma_f32(a1A, b1A, acc11);
    acc00 = wmma_f32(a0B, b0B, acc00);
    acc01 = wmma_f32(a0B, b1B, acc01);
    acc10 = wmma_f32(a1B, b0B, acc10);
    acc11 = wmma_f32(a1B, b1B, acc11);
#pragma unroll
    for (int v = 0; v < 8; ++v) {
      int fr = f0 + v + hi * 8;
      atomicAdd(&G[fr * D + g0 + r],              acc00[v]);
      atomicAdd(&G[fr * D + g0 + 16 + r],         acc01[v]);
      atomicAdd(&G[(fr + 16) * D + g0 + r],       acc10[v]);
      atomicAdd(&G[(fr + 16) * D + g0 + 16 + r],  acc11[v]);
    }
  }
}

// -------------------------------- fused: red = xp@G, combine, bias, BN, ReLU
// OUT = z0*xp + z1*(agg + dis2*xp) - z2*(xp@G) + bias  [; BN; ReLU]
__global__ __launch_bounds__(256) void combine_wmma(
    const float* __restrict__ XP, const float* __restrict__ G,
    const float* __restrict__ AGG, const float* __restrict__ dis2,
    const float* __restrict__ mix, const float* __restrict__ bias,
    const float* __restrict__ bn_g, const float* __restrict__ bn_b,
    const float* __restrict__ bn_m, const float* __restrict__ bn_v,
    float* __restrict__ OUT, int M, int D, int bn_relu)
{
  extern __shared__ float sG[];              // D * D floats, paired-K
  for (int i = threadIdx.x; i < D * D; i += 256) {
    int k = i / D, c = i - k * D;
    sG[(k >> 1) * (2 * D) + c * 2 + (k & 1)] = G[i];
  }
  __syncthreads();
  // softmax(mix) -> zeta, eta, theta (3 elems; redundant per-thread, cached)
  float a0 = mix[0], a1 = mix[1], a2 = mix[2];
  float mx = fmaxf(a0, fmaxf(a1, a2));
  float e0 = expf(a0 - mx), e1 = expf(a1 - mx), e2 = expf(a2 - mx);
  float inv = 1.0f / (e0 + e1 + e2);
  float z0 = e0 * inv, z1 = e1 * inv, z2 = e2 * inv;

  const int wave = threadIdx.x >> 5;
  const int lane = threadIdx.x & 31;
  const int r  = lane & 15;
  const int hi = lane >> 4;
  const int m0 = blockIdx.x * 128 + wave * 16;
  if (m0 >= M) return;
  int ar = m0 + r; if (ar >= M) ar = M - 1;
  const float* Arow = XP + (long)ar * D + 2 * hi;
  for (int n0 = 0; n0 < D; n0 += 64) {
    v8f acc0 = {}, acc1 = {}, acc2 = {}, acc3 = {};
    const float* pb = sG + hi * (2 * D) + (n0 + r) * 2;
    v2f aA  = *(const v2f*)(Arow);
    v2f b0A = *(const v2f*)(pb);
    v2f b1A = *(const v2f*)(pb + 32);
    v2f b2A = *(const v2f*)(pb + 64);
    v2f b3A = *(const v2f*)(pb + 96);
    const float* pbl = pb + 4 * D;
    v2f aB  = *(const v2f*)(Arow + 4);
    v2f b0B = *(const v2f*)(pbl);
    v2f b1B = *(const v2f*)(pbl + 32);
    v2f b2B = *(const v2f*)(pbl + 64);
    v2f b3B = *(const v2f*)(pbl + 96);
    pbl += 4 * D;
    for (int k0 = 8; k0 < D; k0 += 8) {
      acc0 = wmma_f32(aA, b0A, acc0);
      acc1 = wmma_f32(aA, b1A, acc1);
      acc2 = wmma_f32(aA, b2A, acc2);
      acc3 = wmma_f32(aA, b3A, acc3);
      aA  = *(const v2f*)(Arow + k0);
      b0A = *(const v2f*)(pbl);
      b1A = *(const v2f*)(pbl + 32);
      b2A = *(const v2f*)(pbl + 64);
      b3A = *(const v2f*)(pbl + 96);
      pbl += 4 * D;
      acc0 = wmma_f32(aB, b0B, acc0);
      acc1 = wmma_f32(aB, b1B, acc1);
      acc2 = wmma_f32(aB, b2B, acc2);
      acc3 = wmma_f32(aB, b3B, acc3);
      aB  = *(const v2f*)(Arow + k0 + 4);
      b0B = *(const v2f*)(pbl);
      b1B = *(const v2f*)(pbl + 32);
      b2B = *(const v2f*)(pbl + 64);
      b3B = *(const v2f*)(pbl + 96);
      pbl += 4 * D;
    }
    acc0 = wmma_f32(aA, b0A, acc0);
    acc1 = wmma_f32(aA, b1A, acc1);
    acc2 = wmma_f32(aA, b2A, acc2);
    acc3 = wmma_f32(aA, b3A, acc3);
    acc0 = wmma_f32(aB, b0B, acc0);
    acc1 = wmma_f32(aB, b1B, acc1);
    acc2 = wmma_f32(aB, b2B, acc2);
    acc3 = wmma_f32(aB, b3B, acc3);
#pragma unroll
    for (int v = 0; v < 8; ++v) {
      int rr = m0 + v + hi * 8;
      if (rr < M) {
        float d2 = dis2[rr];
#pragma unroll
        for (int q = 0; q < 4; ++q) {
          int cc = n0 + q * 16 + r;
          float red = (q == 0) ? acc0[v] : (q == 1) ? acc1[v]
                    : (q == 2) ? acc2[v] : acc3[v];
          float xpv = XP[(long)rr * D + cc];
          float ag  = AGG[(long)rr * D + cc] + d2 * xpv;   // fold self-loop
          float val = z0 * xpv + z1 * ag - z2 * red + bias[cc];
          if (bn_relu) {
            val = bn_g[cc] * (val - bn_m[cc]) * rsqrtf(bn_v[cc] + BN_EPS) + bn_b[cc];
            val = fmaxf(val, 0.0f);
          }
          OUT[(long)rr * D + cc] = val;
        }
      }
    }
  }
}

// --------------------------------------------------------------------- host
extern "C" void kernel_launch(void* const* d_in, const int* in_sizes, int n_in,
                              void* d_out, int out_size, void* d_ws, size_t ws_size,
                              hipStream_t stream)
{
  (void)in_sizes; (void)n_in; (void)out_size; (void)ws_size;
  const float* x   = (const float*)d_in[0];
  const int*   ei  = (const int*)d_in[1];
  const float* W0  = (const float*)d_in[2];
  const float* b0  = (const float*)d_in[3];
  const float* mw0 = (const float*)d_in[4];
  const float* g0  = (const float*)d_in[5];
  const float* be0 = (const float*)d_in[6];
  const float* rm0 = (const float*)d_in[7];
  const float* rv0 = (const float*)d_in[8];
  const float* W1  = (const float*)d_in[9];
  const float* b1  = (const float*)d_in[10];
  const float* mw1 = (const float*)d_in[11];
  const float* g1  = (const float*)d_in[12];
  const float* be1 = (const float*)d_in[13];
  const float* rm1 = (const float*)d_in[14];
  const float* rv1 = (const float*)d_in[15];
  const float* W2  = (const float*)d_in[16];
  const float* b2  = (const float*)d_in[17];
  const float* mw2 = (const float*)d_in[18];

  const int* rowi = ei;                // edge_index[0]
  const int* coli = ei + N_EDGES;      // edge_index[1]

  // workspace layout (floats)
  float* ws = (float*)d_ws;
  size_t off = 0;
  float* dis  = ws + off; off += N_NODES;
  float* dis2 = ws + off; off += N_NODES;
  float* ew   = ws + off; off += N_EDGES;
  float* xp   = ws + off; off += (size_t)N_NODES * D_H;
  float* agg  = ws + off; off += (size_t)N_NODES * D_H;
  float* hbuf = ws + off; off += (size_t)N_NODES * D_H;
  float* G    = ws + off; off += (size_t)D_H * D_H;

  const dim3 T(256);
  const int MB = (N_NODES + 127) / 128;              // 391 row-blocks
  const int EB = (N_EDGES + 255) / 256;              // 3125
  const int SB = N_EDGES / 8;                        // wave-per-edge blocks

  // ---- cached gcn_norm (computed once, shared by all layers)
  fill_kernel<<<2048, T, 0, stream>>>(dis, 1.0f, (long)N_NODES);   // self loops
  deg_kernel<<<EB, T, 0, stream>>>(coli, dis, N_EDGES);
  norm_kernel<<<(N_NODES + 255) / 256, T, 0, stream>>>(dis, dis2, N_NODES);
  ew_kernel<<<EB, T, 0, stream>>>(rowi, coli, dis, ew, N_EDGES);

  // ---- layer 0: x(128) -> h(128), BN+ReLU
  gemm_wmma<<<MB, T, D_IN * D_H * 4, stream>>>(x, W0, xp, N_NODES, D_IN, D_H);
  fill_kernel<<<2048, T, 0, stream>>>(agg, 0.0f, (long)N_NODES * D_H);
  fill_kernel<<<64, T, 0, stream>>>(G, 0.0f, (long)(D_H * D_H));
  scatter_kernel<<<SB, T, 0, stream>>>(rowi, coli, ew, xp, agg, N_EDGES, D_H);
  gram_wmma<<<MB, T, GRAM_CHUNK * D_H * 4, stream>>>(xp, G, N_NODES, D_H);
  combine_wmma<<<MB, T, D_H * D_H * 4, stream>>>(xp, G, agg, dis2, mw0, b0,
                                                 g0, be0, rm0, rv0,
                                                 hbuf, N_NODES, D_H, 1);

  // ---- layer 1: h(128) -> h(128), BN+ReLU
  gemm_wmma<<<MB, T, D_H * D_H * 4, stream>>>(hbuf, W1, xp, N_NODES, D_H, D_H);
  fill_kernel<<<2048, T, 0, stream>>>(agg, 0.0f, (long)N_NODES * D_H);
  fill_kernel<<<64, T, 0, stream>>>(G, 0.0f, (long)(D_H * D_H));
  scatter_kernel<<<SB, T, 0, stream>>>(rowi, coli, ew, xp, agg, N_EDGES, D_H);
  gram_wmma<<<MB, T, GRAM_CHUNK * D_H * 4, stream>>>(xp, G, N_NODES, D_H);
  combine_wmma<<<MB, T, D_H * D_H * 4, stream>>>(xp, G, agg, dis2, mw1, b1,
                                                 g1, be1, rm1, rv1,
                                                 hbuf, N_NODES, D_H, 1);

  // ---- layer 2: h(128) -> out(64), no BN/ReLU
  gemm_wmma<<<MB, T, D_H * D_OUTF * 4, stream>>>(hbuf, W2, xp, N_NODES, D_H, D_OUTF);
  fill_kernel<<<2048, T, 0, stream>>>(agg, 0.0f, (long)N_NODES * D_OUTF);
  fill_kernel<<<16, T, 0, stream>>>(G, 0.0f, (long)(D_OUTF * D_OUTF));
  scatter_kernel<<<SB, T, 0, stream>>>(rowi, coli, ew, xp, agg, N_EDGES, D_OUTF);
  gram_wmma<<<MB, T, GRAM_CHUNK * D_OUTF * 4, stream>>>(xp, G, N_NODES, D_OUTF);
  combine_wmma<<<MB, T, D_OUTF * D_OUTF * 4, stream>>>(xp, G, agg, dis2, mw2, b2,
                                                       b2, b2, b2, b2,
                                                       (float*)d_out, N_NODES, D_OUTF, 0);
}